// ChamferLossSqrt_45406394253980
// MI455X (gfx1250) — compile-verified
//
#include <hip/hip_runtime.h>
#include <math.h>

typedef float v2f __attribute__((ext_vector_type(2)));
typedef float v8f __attribute__((ext_vector_type(8)));

#define WAVES_PER_BLOCK 8
#define THREADS (WAVES_PER_BLOCK * 32)
#define CHUNK 512   // gt points staged per iteration: 512 * 4 floats = 8KB LDS

// Each wave computes min_j ||p_i - g_j||^2 for its 16 points i over all Mb gts.
// One v_wmma_f32_16x16x4_f32 per 16x16 tile:
//   A[i][k] = (px,py,pz,1), B[k][j] = (-2gx,-2gy,-2gz,|g_j|^2), C = 0
//   => D[i][j] = |g_j|^2 - 2 p_i.g_j ;  |p_i|^2 added after the min-reduction
//   (min_j(pn + x_j) == pn + min_j x_j).
__global__ __launch_bounds__(THREADS)
void chamfer_min_sq_kernel(const float* __restrict__ P,   // [B][Na][3]
                           const float* __restrict__ G,   // [B][Mb][3]
                           float* __restrict__ outMin,    // [B][Na] squared NN dist
                           int Na, int Mb, int blocksPerBatch)
{
    // Pre-transformed B operands: per gt point j -> (-2gx,-2gy,-2gz,gn)
    __shared__ __align__(16) float lds_b[CHUNK * 4];

    const int b       = blockIdx.x / blocksPerBatch;
    const int blkInB  = blockIdx.x % blocksPerBatch;
    const int tid     = threadIdx.x;
    const int wave    = tid >> 5;
    const int lane    = tid & 31;
    const int halfSel = lane >> 4;     // 0: lanes 0-15 (K=0,1), 1: lanes 16-31 (K=2,3)
    const int li      = lane & 15;

    const float* Pb = P + (size_t)b * Na * 3;
    const float* Gb = G + (size_t)b * Mb * 3;

    // Load this wave's 16 points, build the A operand (16x4 f32 layout).
    const int n0 = (blkInB * WAVES_PER_BLOCK + wave) * 16;
    int ip = n0 + li; if (ip >= Na) ip = Na - 1;
    float px = Pb[ip * 3 + 0];
    float py = Pb[ip * 3 + 1];
    float pz = Pb[ip * 3 + 2];
    float pn = px * px + py * py + pz * pz;
    v2f a;
    a.x = halfSel ? pz   : px;   // K=2 : K=0
    a.y = halfSel ? 1.0f : py;   // K=3 : K=1

    v8f acc;
    #pragma unroll
    for (int v = 0; v < 8; ++v) acc[v] = 3.4e38f;

    for (int base = 0; base < Mb; base += CHUNK) {
        __syncthreads();
        // Stage transformed B operands: 2 gt points per thread.
        for (int q = tid; q < CHUNK; q += THREADS) {
            int gidx = base + q;
            if (gidx >= Mb) gidx = Mb - 1;   // clamp: duplicate point is harmless for min
            float gx = Gb[gidx * 3 + 0];
            float gy = Gb[gidx * 3 + 1];
            float gz = Gb[gidx * 3 + 2];
            float gn = gx * gx + gy * gy + gz * gz;
            float4 w;
            w.x = -2.0f * gx; w.y = -2.0f * gy; w.z = -2.0f * gz; w.w = gn;
            ((float4*)lds_b)[q] = w;          // ds_store_b128
        }
        __syncthreads();

        // Per-lane fixed offset into each tile's 16 float4 records.
        const int laneOff = li * 4 + halfSel * 2;
        #pragma unroll 8
        for (int t = 0; t < CHUNK / 16; ++t) {
            v2f bm = *(const v2f*)&lds_b[t * 64 + laneOff];   // ds_load_b64
            v8f c = {};                                       // inline 0 accumulator
            v8f d = __builtin_amdgcn_wmma_f32_16x16x4_f32(
                        false, a, false, bm, (short)0, c, false, false);
            #pragma unroll
            for (int v = 0; v < 8; ++v) acc[v] = fminf(acc[v], d[v]);
        }
    }

    // Row-min: reduce over the 16 columns held across each 16-lane half.
    #pragma unroll
    for (int off = 1; off <= 8; off <<= 1) {
        #pragma unroll
        for (int v = 0; v < 8; ++v) {
            float o = __shfl_xor(acc[v], off, 32);
            acc[v] = fminf(acc[v], o);
        }
    }
    // Lane 0 holds rows 0-7 (VGPR v = row v); lane 16 holds rows 8-15.
    // Fetch |p|^2 for each row from the lane that loaded it, add, clamp, store.
    #pragma unroll
    for (int v = 0; v < 8; ++v) {
        float pnv = __shfl(pn, halfSel * 8 + v, 32);
        acc[v] = fmaxf(acc[v] + pnv, 0.0f);
    }
    if (li == 0) {
        const int rbase = n0 + halfSel * 8;
        float* dst = outMin + (size_t)b * Na + rbase;
        #pragma unroll
        for (int v = 0; v < 8; ++v) {
            if (rbase + v < Na) dst[v] = acc[v];
        }
    }
}

__global__ __launch_bounds__(256)
void batch_reduce_kernel(const float* __restrict__ p2g_sq,  // [B][N]
                         const float* __restrict__ g2p_sq,  // [B][M]
                         float* __restrict__ pb,            // [B] sqrt(mean p2g_sq)
                         float* __restrict__ gb,            // [B]
                         int N, int M)
{
    __shared__ float sp[256], sg[256];
    const int b = blockIdx.x, tid = threadIdx.x;
    float s1 = 0.0f, s2 = 0.0f;
    for (int i = tid; i < N; i += 256) s1 += p2g_sq[(size_t)b * N + i];
    for (int i = tid; i < M; i += 256) s2 += g2p_sq[(size_t)b * M + i];
    sp[tid] = s1; sg[tid] = s2;
    __syncthreads();
    for (int s = 128; s > 0; s >>= 1) {
        if (tid < s) { sp[tid] += sp[tid + s]; sg[tid] += sg[tid + s]; }
        __syncthreads();
    }
    if (tid == 0) {
        pb[b] = sqrtf(sp[0] / (float)N);
        gb[b] = sqrtf(sg[0] / (float)M);
    }
}

__global__ void final_kernel(const float* __restrict__ pb,
                             const float* __restrict__ gb,
                             float* __restrict__ out, int B)
{
    if (threadIdx.x == 0 && blockIdx.x == 0) {
        float mp = 0.0f, mg = 0.0f, ml = 0.0f;
        for (int b = 0; b < B; ++b) {
            mp += pb[b];
            mg += gb[b];
            ml += 0.5f * (pb[b] + gb[b]);
        }
        float inv = 1.0f / (float)B;
        out[0] = ml * inv;   // mean loss
        out[1] = mp * inv;   // mean p2g
        out[2] = mg * inv;   // mean g2p
    }
}

extern "C" void kernel_launch(void* const* d_in, const int* in_sizes, int n_in,
                              void* d_out, int out_size, void* d_ws, size_t ws_size,
                              hipStream_t stream)
{
    const float* points = (const float*)d_in[0];  // [B][N][3]
    const float* gts    = (const float*)d_in[1];  // [B][M][3]
    const int B = 8;
    const int N = in_sizes[0] / (B * 3);
    const int M = in_sizes[1] / (B * 3);

    float* ws     = (float*)d_ws;
    float* p2g_sq = ws;                           // B*N floats
    float* g2p_sq = p2g_sq + (size_t)B * N;       // B*M floats
    float* pb     = g2p_sq + (size_t)B * M;       // B floats
    float* gb     = pb + B;                       // B floats

    const int ptsPerBlock = WAVES_PER_BLOCK * 16;
    const int bpbN = (N + ptsPerBlock - 1) / ptsPerBlock;
    const int bpbM = (M + ptsPerBlock - 1) / ptsPerBlock;

    chamfer_min_sq_kernel<<<B * bpbN, THREADS, 0, stream>>>(points, gts, p2g_sq, N, M, bpbN);
    chamfer_min_sq_kernel<<<B * bpbM, THREADS, 0, stream>>>(gts, points, g2p_sq, M, N, bpbM);
    batch_reduce_kernel<<<B, 256, 0, stream>>>(p2g_sq, g2p_sq, pb, gb, N, M);
    final_kernel<<<1, 32, 0, stream>>>(pb, gb, (float*)d_out, B);
}